// LightweightUserItemGNN_14113262535218
// MI455X (gfx1250) — compile-verified
//
#include <hip/hip_runtime.h>

typedef __attribute__((ext_vector_type(2))) float v2f;
typedef __attribute__((ext_vector_type(8))) float v8f;

#define HD 64
#define WPB 8   // waves per block (256 threads, wave32)

static __device__ __forceinline__ v8f wmma_f32_16x16x4(v2f a, v2f b, v8f c) {
    // D = A(16x4) * B(4x16) + C(16x16), all f32, wave32
    return __builtin_amdgcn_wmma_f32_16x16x4_f32(false, a, false, b, (short)0, c, false, false);
}

static __device__ __forceinline__ void atomAddF(float* p, float v) {
    unsafeAtomicAdd(p, v);   // global_atomic_add_f32, no return
}

// ---------------- utility kernels ----------------

__global__ void zero_f4(float* __restrict__ p, long long n4) {
    long long i = (long long)blockIdx.x * blockDim.x + threadIdx.x;
    long long stride = (long long)gridDim.x * blockDim.x;
    float4 z = make_float4(0.f, 0.f, 0.f, 0.f);
    for (; i < n4; i += stride) ((float4*)p)[i] = z;
}

__global__ void degree_kernel(const int* __restrict__ src, const int* __restrict__ dst,
                              float* __restrict__ degu, float* __restrict__ degi, int E) {
    int i = blockIdx.x * blockDim.x + threadIdx.x;
    int stride = gridDim.x * blockDim.x;
    for (; i < E; i += stride) {
        atomAddF(&degu[src[i]], 1.0f);
        atomAddF(&degi[dst[i]], 1.0f);
    }
}

__global__ void invdeg_kernel(float* __restrict__ d, int n) {
    int i = blockIdx.x * blockDim.x + threadIdx.x;
    if (i < n) d[i] = 1.0f / fmaxf(d[i], 1.0f);
}

// Fused bidirectional edge scatter: accum_u[src] += xi[dst], accum_i[dst] += xu[src].
// 16 lanes per edge, float4 per lane -> 256B coalesced gathers, f32 atomics to L2.
__global__ void scatter_kernel(const int* __restrict__ src, const int* __restrict__ dst,
                               const float* __restrict__ xu, const float* __restrict__ xi,
                               float* __restrict__ au, float* __restrict__ ai, long long total) {
    long long idx = (long long)blockIdx.x * blockDim.x + threadIdx.x;
    long long stride = (long long)gridDim.x * blockDim.x;
    for (; idx < total; idx += stride) {
        long long e = idx >> 4;
        int c = (int)(idx & 15) << 2;
        int s = src[e];
        int d = dst[e];
        float4 vi = *(const float4*)(xi + (size_t)d * HD + c);
        float* pu = au + (size_t)s * HD + c;
        atomAddF(pu + 0, vi.x);
        atomAddF(pu + 1, vi.y);
        atomAddF(pu + 2, vi.z);
        atomAddF(pu + 3, vi.w);
        float4 vu = *(const float4*)(xu + (size_t)s * HD + c);
        float* pi = ai + (size_t)d * HD + c;
        atomAddF(pi + 0, vu.x);
        atomAddF(pi + 1, vu.y);
        atomAddF(pi + 2, vu.z);
        atomAddF(pi + 3, vu.w);
    }
}

// ---------------- node update: out = relu(bn(agg@Wagg + bias + x@Wself)) ----------------
// One wave per 16-row tile. agg = accum * invdeg folded into A-fragment loads.
// accum/out may alias (row-partitioned, no __restrict__ on that pair).
__global__ __launch_bounds__(256) void node_update_kernel(
    const float* __restrict__ xself, const float* accum, const float* __restrict__ invdeg,
    const float* __restrict__ Wagg, const float* __restrict__ Wself,
    const float* __restrict__ bias,
    const float* __restrict__ bng, const float* __restrict__ bnb,
    const float* __restrict__ bnm, const float* __restrict__ bnv,
    float* out, int N)
{
    __shared__ float sWa[HD * HD];
    __shared__ float sWs[HD * HD];
    __shared__ float sScale[HD];
    __shared__ float sShift[HD];
    int t = threadIdx.x;
    for (int i = t; i < HD * HD; i += 256) { sWa[i] = Wagg[i]; sWs[i] = Wself[i]; }
    if (t < HD) {
        float sc = rsqrtf(bnv[t] + 1e-5f) * bng[t];
        sScale[t] = sc;
        sShift[t] = (bias[t] - bnm[t]) * sc + bnb[t];
    }
    __syncthreads();

    int wave = t >> 5, lane = t & 31;
    int tile = blockIdx.x * WPB + wave;
    int row0 = tile * 16;
    if (row0 >= N) return;

    int m = lane & 15;
    int koff = (lane >> 4) << 1;           // lanes 16-31 carry K+2,K+3
    int rowi = row0 + m; if (rowi > N - 1) rowi = N - 1;
    const float* ap = accum + (size_t)rowi * HD;
    const float* sp = xself + (size_t)rowi * HD;
    float idg = invdeg[rowi];

    v8f acc[4] = {v8f{}, v8f{}, v8f{}, v8f{}};
    for (int kb = 0; kb < HD; kb += 4) {
        int k = kb + koff;
        v2f aA, aS;
        aA.x = ap[k] * idg; aA.y = ap[k + 1] * idg;
        aS.x = sp[k];       aS.y = sp[k + 1];
#pragma unroll
        for (int nb = 0; nb < 4; ++nb) {
            int n = nb * 16 + m;
            v2f bA, bS;
            bA.x = sWa[k * HD + n]; bA.y = sWa[(k + 1) * HD + n];
            bS.x = sWs[k * HD + n]; bS.y = sWs[(k + 1) * HD + n];
            acc[nb] = wmma_f32_16x16x4(aA, bA, acc[nb]);
            acc[nb] = wmma_f32_16x16x4(aS, bS, acc[nb]);
        }
    }
#pragma unroll
    for (int nb = 0; nb < 4; ++nb) {
        int n = nb * 16 + m;
        float sc = sScale[n], sh = sShift[n];
#pragma unroll
        for (int r = 0; r < 8; ++r) {
            int mr = (lane < 16) ? r : r + 8;
            if (row0 + mr < N) {
                float v = fmaxf(acc[nb][r] * sc + sh, 0.0f);
                out[(size_t)(row0 + mr) * HD + n] = v;
            }
        }
    }
}

// ---------------- final MLP: relu([u|i] @ fc1 + b1) @ fc2 + b2 ----------------
__global__ __launch_bounds__(256) void mlp_kernel(
    const int* __restrict__ eli,
    const float* __restrict__ xu, const float* __restrict__ xi,
    const float* __restrict__ fc1w, const float* __restrict__ fc1b,
    const float* __restrict__ fc2w, const float* __restrict__ fc2b,
    float* __restrict__ out, int Pn)
{
    __shared__ float sW1[2 * HD * HD];       // 128x64
    __shared__ float sB1[HD];
    __shared__ float sW2[HD * 4];
    __shared__ float sB2[4];
    __shared__ float sHid[WPB][16 * HD];     // per-wave hidden tile
    int t = threadIdx.x;
    for (int i = t; i < 2 * HD * HD; i += 256) sW1[i] = fc1w[i];
    for (int i = t; i < HD * 4; i += 256) sW2[i] = fc2w[i];
    if (t < HD) sB1[t] = fc1b[t];
    if (t < 4) sB2[t] = fc2b[t];
    __syncthreads();

    int wave = t >> 5, lane = t & 31;
    int tile = blockIdx.x * WPB + wave;
    int row0 = tile * 16;
    if (row0 >= Pn) return;

    int m = lane & 15;
    int koff = (lane >> 4) << 1;
    int row = row0 + m; if (row > Pn - 1) row = Pn - 1;
    const float* up = xu + (size_t)eli[row] * HD;
    const float* ip = xi + (size_t)eli[Pn + row] * HD;

    v8f acc[4] = {v8f{}, v8f{}, v8f{}, v8f{}};
    for (int kb = 0; kb < HD; kb += 4) {          // user half of concat
        int k = kb + koff;
        v2f a; a.x = up[k]; a.y = up[k + 1];
#pragma unroll
        for (int nb = 0; nb < 4; ++nb) {
            int n = nb * 16 + m;
            v2f b; b.x = sW1[k * HD + n]; b.y = sW1[(k + 1) * HD + n];
            acc[nb] = wmma_f32_16x16x4(a, b, acc[nb]);
        }
    }
    for (int kb = 0; kb < HD; kb += 4) {          // item half of concat
        int k = kb + koff;
        v2f a; a.x = ip[k]; a.y = ip[k + 1];
#pragma unroll
        for (int nb = 0; nb < 4; ++nb) {
            int n = nb * 16 + m;
            v2f b; b.x = sW1[(HD + k) * HD + n]; b.y = sW1[(HD + k + 1) * HD + n];
            acc[nb] = wmma_f32_16x16x4(a, b, acc[nb]);
        }
    }

    float* Hd = sHid[wave];
#pragma unroll
    for (int nb = 0; nb < 4; ++nb) {
        int n = nb * 16 + m;
        float b1 = sB1[n];
#pragma unroll
        for (int r = 0; r < 8; ++r) {
            int mr = (lane < 16) ? r : r + 8;
            Hd[mr * HD + n] = fmaxf(acc[nb][r] + b1, 0.0f);
        }
    }
    asm volatile("s_wait_dscnt 0" ::: "memory");  // same-wave LDS RAW fence

    int orow = lane >> 1;
    int oc = (lane & 1) * 2;
    float s0 = sB2[oc], s1 = sB2[oc + 1];
    for (int k = 0; k < HD; ++k) {
        float h = Hd[orow * HD + k];
        s0 += h * sW2[k * 4 + oc];
        s1 += h * sW2[k * 4 + oc + 1];
    }
    if (row0 + orow < Pn) {
        out[(size_t)(row0 + orow) * 4 + oc]     = s0;
        out[(size_t)(row0 + orow) * 4 + oc + 1] = s1;
    }
}

// ---------------- host orchestration ----------------

extern "C" void kernel_launch(void* const* d_in, const int* in_sizes, int n_in,
                              void* d_out, int out_size, void* d_ws, size_t ws_size,
                              hipStream_t stream) {
    const int*   edge_index = (const int*)d_in[0];
    const int*   eli        = (const int*)d_in[1];
    const float* user_emb   = (const float*)d_in[2];
    const float* item_emb   = (const float*)d_in[3];
    const float* u_ll_w = (const float*)d_in[4];
    const float* u_ll_b = (const float*)d_in[5];
    const float* u_lr_w = (const float*)d_in[6];
    const float* i_ll_w = (const float*)d_in[7];
    const float* i_ll_b = (const float*)d_in[8];
    const float* i_lr_w = (const float*)d_in[9];
    const float* u_bn_g = (const float*)d_in[10];
    const float* u_bn_b = (const float*)d_in[11];
    const float* u_bn_m = (const float*)d_in[12];
    const float* u_bn_v = (const float*)d_in[13];
    const float* i_bn_g = (const float*)d_in[14];
    const float* i_bn_b = (const float*)d_in[15];
    const float* i_bn_m = (const float*)d_in[16];
    const float* i_bn_v = (const float*)d_in[17];
    const float* fc1w = (const float*)d_in[18];
    const float* fc1b = (const float*)d_in[19];
    const float* fc2w = (const float*)d_in[20];
    const float* fc2b = (const float*)d_in[21];

    const int E  = in_sizes[0] / 2;
    const int Pn = in_sizes[1] / 2;
    const int NU = in_sizes[2] / HD;
    const int NI = in_sizes[3] / HD;
    const int* src = edge_index;
    const int* dst = edge_index + E;

    float* ws = (float*)d_ws;
    size_t NFu = (size_t)NU * HD, NFi = (size_t)NI * HD;
    float* A = ws;            // user accum -> user_x after layer 0
    float* B = A + NFu;       // item accum -> item_x after layer 0
    float* C = B + NFi;       // user accum -> user_x after layer 1
    float* D = C + NFu;       // item accum -> item_x after layer 1
    float* degu = D + NFi;    // degree -> invdeg (users)
    float* degi = degu + NU;  // degree -> invdeg (items)

    dim3 blk(256);
    auto nb = [](long long n, int b) { return (unsigned)((n + b - 1) / b); };

    // degrees (layer-invariant)
    zero_f4<<<nb(NU / 4, 256), blk, 0, stream>>>(degu, NU / 4);
    zero_f4<<<nb(NI / 4, 256), blk, 0, stream>>>(degi, NI / 4);
    degree_kernel<<<nb(E, 256), blk, 0, stream>>>(src, dst, degu, degi, E);
    invdeg_kernel<<<nb(NU, 256), blk, 0, stream>>>(degu, NU);
    invdeg_kernel<<<nb(NI, 256), blk, 0, stream>>>(degi, NI);

    const long long tot = (long long)E * 16;
    const unsigned sblocks = nb(tot, 256);
    const unsigned ublocks = nb((NU + 15) / 16, WPB);
    const unsigned iblocks = nb((NI + 15) / 16, WPB);

    // ---- layer 0 (aggregates use input embeddings) ----
    zero_f4<<<nb((long long)NFu / 4, 256), blk, 0, stream>>>(A, NFu / 4);
    zero_f4<<<nb((long long)NFi / 4, 256), blk, 0, stream>>>(B, NFi / 4);
    scatter_kernel<<<sblocks, blk, 0, stream>>>(src, dst, user_emb, item_emb, A, B, tot);
    node_update_kernel<<<ublocks, blk, 0, stream>>>(user_emb, A, degu,
        u_ll_w, u_lr_w, u_ll_b, u_bn_g, u_bn_b, u_bn_m, u_bn_v, A, NU);
    node_update_kernel<<<iblocks, blk, 0, stream>>>(item_emb, B, degi,
        i_ll_w, i_lr_w, i_ll_b, i_bn_g, i_bn_b, i_bn_m, i_bn_v, B, NI);

    // ---- layer 1 (aggregates use layer-0 outputs A, B) ----
    zero_f4<<<nb((long long)NFu / 4, 256), blk, 0, stream>>>(C, NFu / 4);
    zero_f4<<<nb((long long)NFi / 4, 256), blk, 0, stream>>>(D, NFi / 4);
    scatter_kernel<<<sblocks, blk, 0, stream>>>(src, dst, A, B, C, D, tot);
    node_update_kernel<<<ublocks, blk, 0, stream>>>(A, C, degu,
        u_ll_w + HD * HD, u_lr_w + HD * HD, u_ll_b + HD,
        u_bn_g + HD, u_bn_b + HD, u_bn_m + HD, u_bn_v + HD, C, NU);
    node_update_kernel<<<iblocks, blk, 0, stream>>>(B, D, degi,
        i_ll_w + HD * HD, i_lr_w + HD * HD, i_ll_b + HD,
        i_bn_g + HD, i_bn_b + HD, i_bn_m + HD, i_bn_v + HD, D, NI);

    // ---- edge-pair MLP head ----
    mlp_kernel<<<nb((Pn + 15) / 16, WPB), blk, 0, stream>>>(
        eli, C, D, fc1w, fc1b, fc2w, fc2b, (float*)d_out, Pn);
}